// Global_62242666054207
// MI455X (gfx1250) — compile-verified
//
#include <hip/hip_runtime.h>
#include <math.h>

typedef __attribute__((ext_vector_type(2))) float v2f;
typedef __attribute__((ext_vector_type(8))) float v8f;

#define WAVES 4  // 4 wave32 per block

// ---------------------------------------------------------------------------
// Generic fp32 WMMA GEMM:  C[m,n] = alpha * sum_k op(A)[m,k]*op(B)[k,n] + bias[n]
// AMODE 0: A stored [M,K] (row-major, lda)   AMODE 1: A stored [K,M] (op(A)=Aᵀ)
// BMODE 0: B stored [K,N] (ldb)              BMODE 1: B stored [N,K] (op(B)=Bᵀ)
// Each wave owns a (16*MT) x (16*NT) tile of C; K consumed 4 per step via
// v_wmma_f32_16x16x4_f32 with a 1-deep software pipeline (prefetch k+4 while
// the k fragments are consumed) so s_wait_loadcnt never drains to zero on the
// critical path.  Fragment layouts per CDNA5 ISA 7.12.2.
// ---------------------------------------------------------------------------
template<int AMODE, int BMODE, int MT, int NT>
__global__ __launch_bounds__(128) void wmma_gemm(
    const float* __restrict__ A, const float* __restrict__ Bm,
    float* __restrict__ C, const float* __restrict__ bias,
    int M, int K, int lda, int ldb, int ldc,
    long sA, long sB, long sC, float alpha, int relu)
{
  const int lane = threadIdx.x & 31;
  const int wave = threadIdx.x >> 5;
  const int hl   = lane >> 4;      // lane half: k pair {0,1} vs {2,3}
  const int li   = lane & 15;
  const int m0 = (blockIdx.x * WAVES + wave) * (16 * MT);
  const int n0 = blockIdx.y * (16 * NT);
  if (m0 >= M) return;             // uniform per wave: EXEC stays all-ones
  A  += (long)blockIdx.z * sA;
  Bm += (long)blockIdx.z * sB;
  C  += (long)blockIdx.z * sC;

  v8f acc[MT * NT] = {};
  v2f ac[MT], bc[NT];

  // fragment loaders -----------------------------------------------------
  auto loadA = [&](int k0, v2f* a) {
#pragma unroll
    for (int i = 0; i < MT; ++i) {
      if (AMODE == 0) {            // A[m][k]: contiguous pair
        const float* p = A + (long)(m0 + 16 * i + li) * lda + (k0 + 2 * hl);
        a[i].x = p[0]; a[i].y = p[1];
      } else {                     // op(A)[m][k] = A[k][m]
        const float* p = A + (long)(k0 + 2 * hl) * lda + (m0 + 16 * i + li);
        a[i].x = p[0]; a[i].y = p[lda];
      }
    }
  };
  auto loadB = [&](int k0, v2f* b) {
#pragma unroll
    for (int t = 0; t < NT; ++t) {
      const int n = n0 + 16 * t + li;
      if (BMODE == 0) {            // B[k][n]
        const float* p = Bm + (long)(k0 + 2 * hl) * ldb + n;
        b[t].x = p[0]; b[t].y = p[ldb];
      } else {                     // op(B)[k][n] = B[n][k]: contiguous pair
        const float* p = Bm + (long)n * ldb + (k0 + 2 * hl);
        b[t].x = p[0]; b[t].y = p[1];
      }
    }
  };
  auto mma = [&](const v2f* a, const v2f* b) {
#pragma unroll
    for (int i = 0; i < MT; ++i)
#pragma unroll
      for (int t = 0; t < NT; ++t)
        acc[i * NT + t] = __builtin_amdgcn_wmma_f32_16x16x4_f32(
            false, a[i], false, b[t], (short)0, acc[i * NT + t], false, false);
  };

  // 1-deep pipelined K loop ---------------------------------------------
  loadA(0, ac); loadB(0, bc);
  for (int k0 = 0; k0 + 4 < K; k0 += 4) {
    v2f an[MT], bn[NT];
    loadA(k0 + 4, an);             // next fragments in flight...
    loadB(k0 + 4, bn);
    mma(ac, bc);                   // ...while current ones are consumed
#pragma unroll
    for (int i = 0; i < MT; ++i) ac[i] = an[i];
#pragma unroll
    for (int t = 0; t < NT; ++t) bc[t] = bn[t];
  }
  mma(ac, bc);                     // epilogue step

  // C/D layout: VGPR r, lanes 0-15 -> row r, lanes 16-31 -> row r+8; col = li
#pragma unroll
  for (int i = 0; i < MT; ++i)
#pragma unroll
    for (int t = 0; t < NT; ++t) {
      const int n = n0 + 16 * t + li;
      const float bv = bias ? bias[n] : 0.0f;
#pragma unroll
      for (int r = 0; r < 8; ++r) {
        const int row = m0 + 16 * i + r + 8 * hl;
        float v = acc[i * NT + t][r] * alpha + bv;
        if (relu) v = fmaxf(v, 0.0f);
        C[(long)row * ldc + n] = v;
      }
    }
}

// ---------------------------------------------------------------------------
// Small elementwise / reduction kernels
// ---------------------------------------------------------------------------
__global__ void sum_over_b(const float* __restrict__ q, float* __restrict__ xs,
                           int per, int Bsz)
{
  int i = blockIdx.x * 256 + threadIdx.x;
  if (i >= per) return;
  float s = 0.f;
  for (int b = 0; b < Bsz; ++b) s += q[(long)b * per + i];
  xs[i] = s;
}

// effective bias for the folded R-path: Bsz*(ffb1 @ ffw2ᵀ + ffb2)
__global__ void bias_eff(const float* __restrict__ b1, const float* __restrict__ W2,
                         const float* __restrict__ b2, float* __restrict__ be, int Bsz)
{
  int o = threadIdx.x;  // 64
  float s = 0.f;
  for (int k = 0; k < 384; ++k) s += b1[k] * W2[o * 384 + k];
  be[o] = (float)Bsz * (s + b2[o]);
}

// in-place row softmax; optionally add identity (relu(softmax)==softmax since >0)
__global__ void softmax_rows(float* __restrict__ Mx, int Ncols, int addEye)
{
  __shared__ float red[256];
  const int row = blockIdx.x, tid = threadIdx.x;
  float* r = Mx + (long)row * Ncols;
  float mx = -3.402823466e38f;
  for (int j = tid; j < Ncols; j += 256) mx = fmaxf(mx, r[j]);
  red[tid] = mx; __syncthreads();
  for (int s = 128; s > 0; s >>= 1) { if (tid < s) red[tid] = fmaxf(red[tid], red[tid + s]); __syncthreads(); }
  mx = red[0]; __syncthreads();
  float sum = 0.f;
  for (int j = tid; j < Ncols; j += 256) { float e = __expf(r[j] - mx); r[j] = e; sum += e; }
  red[tid] = sum; __syncthreads();
  for (int s = 128; s > 0; s >>= 1) { if (tid < s) red[tid] += red[tid + s]; __syncthreads(); }
  const float inv = 1.f / red[0];
  for (int j = tid; j < Ncols; j += 256) {
    float v = r[j] * inv;
    if (addEye && j == row) v += 1.f;
    r[j] = v;
  }
}

__global__ void copy_q_cat(const float* __restrict__ q, float* __restrict__ cat, int total)
{
  int i = blockIdx.x * 256 + threadIdx.x;
  if (i >= total) return;
  int c = i & 63, bn = i >> 6;
  cat[(long)bn * 192 + c] = q[i];
}

__global__ void fuse_gate(const float* __restrict__ xd, const float* __restrict__ xf,
                          float* __restrict__ o, int total)
{
  int i = blockIdx.x * 256 + threadIdx.x;
  if (i >= total) return;
  float t = tanhf(xd[i]);
  float s = 1.f / (1.f + __expf(-xf[i]));
  o[i] = t * s;
}

// out_row = LayerNorm(a_row + b_row) * g + beta   (row length 64, one block/row)
__global__ void add_ln(const float* __restrict__ a, const float* __restrict__ b,
                       const float* __restrict__ g, const float* __restrict__ be,
                       float* __restrict__ out)
{
  __shared__ float red[64];
  const int row = blockIdx.x, t = threadIdx.x;
  float x = a[(long)row * 64 + t] + b[(long)row * 64 + t];
  red[t] = x; __syncthreads();
  for (int s = 32; s > 0; s >>= 1) { if (t < s) red[t] += red[t + s]; __syncthreads(); }
  float mean = red[0] * (1.f / 64.f); __syncthreads();
  float d = x - mean;
  red[t] = d * d; __syncthreads();
  for (int s = 32; s > 0; s >>= 1) { if (t < s) red[t] += red[t + s]; __syncthreads(); }
  float var = red[0] * (1.f / 64.f);
  out[(long)row * 64 + t] = d * rsqrtf(var + 1e-5f) * g[t] + be[t];
}

// ---------------------------------------------------------------------------
extern "C" void kernel_launch(void* const* d_in, const int* in_sizes, int n_in,
                              void* d_out, int out_size, void* d_ws, size_t ws_size,
                              hipStream_t stream)
{
  (void)in_sizes; (void)n_in; (void)out_size; (void)ws_size;
  const int B = 8, N = 2048;
  const float* query = (const float*)d_in[0];
  const float* adj   = (const float*)d_in[1];
  const float* WF    = (const float*)d_in[2];
  const float* bF    = (const float*)d_in[3];
  const float* WD    = (const float*)d_in[4];
  const float* bD    = (const float*)d_in[5];
  const float* ffw1  = (const float*)d_in[6];
  const float* ffb1  = (const float*)d_in[7];
  const float* ffw2  = (const float*)d_in[8];
  const float* ffb2  = (const float*)d_in[9];
  const float* wq    = (const float*)d_in[10];
  const float* wk    = (const float*)d_in[11];
  const float* wv    = (const float*)d_in[12];
  const float* wo    = (const float*)d_in[13];
  const float* bo    = (const float*)d_in[14];
  const float* g2    = (const float*)d_in[15];
  const float* be2   = (const float*)d_in[16];
  const float* g3    = (const float*)d_in[17];
  const float* be3   = (const float*)d_in[18];
  const float* f1w   = (const float*)d_in[19];
  const float* f1b   = (const float*)d_in[20];
  const float* f2w   = (const float*)d_in[21];
  const float* f2b   = (const float*)d_in[22];
  float* out = (float*)d_out;

  float* ws = (float*)d_ws;
  const long NN = (long)N * N;         // 4,194,304
  const long BN64 = (long)B * N * 64;  // 1,048,576
  long o = 0;
  float* Rm   = ws + o; o += NN;             // dynamic adjacency (in-place softmax+I)
  float* Sb   = ws + o; o += NN;             // attention scores / FFN hidden (16384x256)
  float* cat  = ws + o; o += (long)B*N*192;  // [q|h1|h2] concat; later q/k/v proj
  float* tmp  = ws + o; o += (long)N*384;
  float* Xs   = ws + o; o += (long)N*64;
  float* Rmat = ws + o; o += (long)N*64;
  float* bEff = ws + o; o += 64;
  float* XF   = ws + o; o += BN64;
  float* XD   = ws + o; o += BN64;
  float* XDF  = ws + o; o += BN64;
  float* ctx  = ws + o; o += BN64;
  float* xres = ws + o; o += BN64;

  dim3 blk(128);
  // per-wave output tile = 32x64 (MT=2, NT=4)  ->  block covers 128 rows

  // ---- dynamic adjacency path (linear -> fold batch sum first) ----
  sum_over_b<<<(N * 64 + 255) / 256, 256, 0, stream>>>(query, Xs, N * 64, B);
  // tmp[2048,384] = Xs @ ffw1ᵀ
  wmma_gemm<0,1,2,4><<<dim3(16, 6, 1), blk, 0, stream>>>(
      Xs, ffw1, tmp, nullptr, 2048, 64, 64, 64, 384, 0, 0, 0, 1.0f, 0);
  bias_eff<<<1, 64, 0, stream>>>(ffb1, ffw2, ffb2, bEff, B);
  // Rmat[2048,64] = tmp @ ffw2ᵀ + bEff
  wmma_gemm<0,1,2,4><<<dim3(16, 1, 1), blk, 0, stream>>>(
      tmp, ffw2, Rmat, bEff, 2048, 384, 384, 384, 64, 0, 0, 0, 1.0f, 0);
  // Rm = Rmat @ Rmatᵀ
  wmma_gemm<0,1,2,4><<<dim3(16, 32, 1), blk, 0, stream>>>(
      Rmat, Rmat, Rm, nullptr, 2048, 64, 64, 64, 2048, 0, 0, 0, 1.0f, 0);
  softmax_rows<<<2048, 256, 0, stream>>>(Rm, 2048, 1);  // relu(softmax)+I

  // ---- two GCN branches: hops (TN, batched over B) + 1x1 conv ----
  copy_q_cat<<<(int)((BN64 + 255) / 256), 256, 0, stream>>>(query, cat, (int)BN64);
  const float* Amats[2] = { adj, Rm };
  const float* Wc[2]    = { WF, WD };
  const float* bc[2]    = { bF, bD };
  float*       Xo[2]    = { XF, XD };
  for (int br = 0; br < 2; ++br) {
    // h1[b,m,c] = sum_n A[n,m] q[b,n,c]  -> cat[:,:,64:128]
    wmma_gemm<1,0,2,4><<<dim3(16, 1, 8), blk, 0, stream>>>(
        Amats[br], query, cat + 64, nullptr, 2048, 2048, 2048, 64, 192,
        0, (long)N * 64, (long)N * 192, 1.0f, 0);
    // h2 = Aᵀ h1 -> cat[:,:,128:192]
    wmma_gemm<1,0,2,4><<<dim3(16, 1, 8), blk, 0, stream>>>(
        Amats[br], cat + 64, cat + 128, nullptr, 2048, 2048, 2048, 192, 192,
        0, (long)N * 192, (long)N * 192, 1.0f, 0);
    // X = cat @ Wᵀ + b   (M = B*N, K = 192)
    wmma_gemm<0,1,2,4><<<dim3(128, 1, 1), blk, 0, stream>>>(
        cat, Wc[br], Xo[br], bc[br], 16384, 192, 192, 192, 64, 0, 0, 0, 1.0f, 0);
  }

  // ---- gated fusion ----
  fuse_gate<<<(int)((BN64 + 255) / 256), 256, 0, stream>>>(XD, XF, XDF, (int)BN64);

  // ---- q/k/v projections (reuse cat region) ----
  float* qp = cat; float* kp = cat + BN64; float* vp = cat + 2 * BN64;
  wmma_gemm<0,1,2,4><<<dim3(128, 1, 1), blk, 0, stream>>>(XDF, wq, qp, nullptr, 16384, 64, 64, 64, 64, 0, 0, 0, 1.0f, 0);
  wmma_gemm<0,1,2,4><<<dim3(128, 1, 1), blk, 0, stream>>>(XDF, wk, kp, nullptr, 16384, 64, 64, 64, 64, 0, 0, 0, 1.0f, 0);
  wmma_gemm<0,1,2,4><<<dim3(128, 1, 1), blk, 0, stream>>>(XDF, wv, vp, nullptr, 16384, 64, 64, 64, 64, 0, 0, 0, 1.0f, 0);

  // ---- spatial multi-head attention over N (per (b,h), L2-resident scratch) ----
  for (int bh = 0; bh < B * 4; ++bh) {
    const int b = bh >> 2, h = bh & 3;
    const long off = (long)b * N * 64 + (long)h * 16;
    // scores = q khᵀ / sqrt(16)
    wmma_gemm<0,1,2,4><<<dim3(16, 32, 1), blk, 0, stream>>>(
        qp + off, kp + off, Sb, nullptr, 2048, 16, 64, 64, 2048, 0, 0, 0, 0.25f, 0);
    softmax_rows<<<2048, 256, 0, stream>>>(Sb, 2048, 0);
    // ctx = attn @ v   (NN)
    wmma_gemm<0,0,2,1><<<dim3(16, 1, 1), blk, 0, stream>>>(
        Sb, vp + off, ctx + off, nullptr, 2048, 2048, 2048, 64, 64, 0, 0, 0, 1.0f, 0);
  }

  // ---- output proj + residual LN + FFN + final LN ----
  wmma_gemm<0,1,2,4><<<dim3(128, 1, 1), blk, 0, stream>>>(ctx, wo, xres, bo, 16384, 64, 64, 64, 64, 0, 0, 0, 1.0f, 0);
  add_ln<<<16384, 64, 0, stream>>>(xres, XDF, g2, be2, xres);
  // hidden = relu(x @ f1wᵀ + f1b)  -> Sb (16384 x 256)
  wmma_gemm<0,1,2,4><<<dim3(128, 4, 1), blk, 0, stream>>>(xres, f1w, Sb, f1b, 16384, 64, 64, 64, 256, 0, 0, 0, 1.0f, 1);
  // ffn = hidden @ f2wᵀ + f2b -> ctx (reuse)
  wmma_gemm<0,1,2,4><<<dim3(128, 1, 1), blk, 0, stream>>>(Sb, f2w, ctx, f2b, 16384, 256, 256, 256, 64, 0, 0, 0, 1.0f, 0);
  add_ln<<<16384, 64, 0, stream>>>(ctx, xres, g3, be3, out);
}